// ExportableMGNetPostProcessing_15564961480716
// MI455X (gfx1250) — compile-verified
//
#include <hip/hip_runtime.h>
#include <hip/hip_bf16.h>
#include <stdint.h>

typedef __attribute__((ext_vector_type(2))) float v2f;
typedef __attribute__((ext_vector_type(8))) float v8f;

#define H_   1024
#define W_   2048
#define HW_  (H_*W_)
#define CAP_ 16384

// workspace byte offsets
#define OFF_CAND_CNT   0
#define OFF_GROUND_CNT 4
#define OFF_PREFIX     8
#define OFF_KREMAIN    12
#define OFF_ANYVALID   16
#define OFF_BINS       32            // 1024 x u32 (4 passes x 256 bins)
#define OFF_CAND       8192          // CAP_ x u64
#define OFF_CENTERS    (160*1024)    // negy2[32], negx2[32], cc[32] floats
#define OFF_KEYS       (256*1024)    // HW_ x u32
#define ZERO_WORDS     ((OFF_CAND + CAP_*8)/4)

// ---------------- k0: zero counters / bins / candidate region ----------------
__global__ void k0_zero(unsigned* ws32) {
    int i = blockIdx.x * blockDim.x + threadIdx.x;
    int n = ZERO_WORDS;
    for (; i < n; i += gridDim.x * blockDim.x) ws32[i] = 0u;
}

// ---------------- k1: compact positive heatmap entries ----------------
__global__ void k1_cand(const float* __restrict__ heat, char* ws) {
    unsigned* cnt = (unsigned*)(ws + OFF_CAND_CNT);
    unsigned long long* cand = (unsigned long long*)(ws + OFF_CAND);
    int i = blockIdx.x * blockDim.x + threadIdx.x;
    for (; i < HW_; i += gridDim.x * blockDim.x) {
        float v = heat[i];
        if (v > 0.0f) {
            unsigned pos = atomicAdd(cnt, 1u);
            if (pos < CAP_) {
                unsigned long long key =
                    ((unsigned long long)__float_as_uint(v) << 32) |
                    (unsigned long long)(~(unsigned)i);
                cand[pos] = key;  // larger value wins; tie -> smaller index wins
            }
        }
    }
}

// ---------------- k2: single-block top-32 extraction ----------------
__global__ void k2_top(char* ws) {
    __shared__ unsigned long long red[256];
    __shared__ unsigned long long bestsh;
    unsigned n = *(unsigned*)(ws + OFF_CAND_CNT);
    if (n > CAP_) n = CAP_;
    unsigned long long* cand = (unsigned long long*)(ws + OFF_CAND);
    float* negy2 = (float*)(ws + OFF_CENTERS);
    float* negx2 = negy2 + 32;
    float* cc    = negy2 + 64;
    int tid = threadIdx.x;
    int nv = 0;
    for (int k = 0; k < 32; ++k) {
        unsigned long long local = 0ull;
        for (unsigned i = tid; i < n; i += 256) {
            unsigned long long c = cand[i];
            if (c > local) local = c;
        }
        red[tid] = local;
        __syncthreads();
        for (int s = 128; s > 0; s >>= 1) {
            if (tid < s && red[tid + s] > red[tid]) red[tid] = red[tid + s];
            __syncthreads();
        }
        if (tid == 0) bestsh = red[0];
        __syncthreads();
        unsigned long long best = bestsh;
        if (best) {  // remove chosen entry (unique: low bits encode index)
            for (unsigned i = tid; i < n; i += 256)
                if (cand[i] == best) cand[i] = 0ull;
        }
        if (tid == 0) {
            if (best) {
                unsigned idx = ~(unsigned)(best & 0xFFFFFFFFull);
                float cy = (float)(idx >> 11);       // idx / 2048
                float cx = (float)(idx & 2047);      // idx % 2048
                negy2[k] = -2.0f * cy;
                negx2[k] = -2.0f * cx;
                cc[k]    = cy * cy + cx * cx;
                ++nv;
            } else {
                negy2[k] = 0.0f; negx2[k] = 0.0f; cc[k] = 1e30f;  // invalid center
            }
        }
        __syncthreads();
    }
    if (tid == 0) *(int*)(ws + OFF_ANYVALID) = (nv > 0) ? 1 : 0;
}

// ---------------- k3: WMMA nearest-center instance assignment ----------------
// D = A(16 centers x [ -2cy, -2cx, |c|^2, 0 ]) x B(4 x 16 pixels [ys, xs, 1, 0])
// per-pixel |p|^2 term is constant across centers -> argmin unaffected.
__global__ void k3_inst(const int* __restrict__ sem, const float* __restrict__ off,
                        const char* __restrict__ ws, float* out) {
    const float* negy2 = (const float*)(ws + OFF_CENTERS);
    const float* negx2 = negy2 + 32;
    const float* cc    = negy2 + 64;
    const int anyv = *(const int*)(ws + OFF_ANYVALID);
    int* outSem = (int*)out;

    int lane = threadIdx.x & 31;
    int l16  = lane & 15;
    bool lo  = lane < 16;
    int wavesPerBlock = blockDim.x >> 5;
    int wid = blockIdx.x * wavesPerBlock + (threadIdx.x >> 5);
    int nW  = gridDim.x * wavesPerBlock;

    // A layout (32-bit 16x4): lanes 0-15 hold (K0,K1) for row M=lane,
    // lanes 16-31 hold (K2,K3) for row M=lane-16.
    v2f a0, a1;
    a0.x = lo ? negy2[l16]      : cc[l16];
    a0.y = lo ? negx2[l16]      : 0.0f;
    a1.x = lo ? negy2[l16 + 16] : cc[l16 + 16];
    a1.y = lo ? negx2[l16 + 16] : 0.0f;

    const int nGroups = HW_ / 16;
    for (int g = wid; g < nGroups; g += nW) {
        int p = g * 16 + l16;
        int y = p >> 11;
        int x = p & 2047;
        float dy = off[p];
        float dx = off[HW_ + p];
        float ys = (float)y + dy;
        float xs = (float)x + dx;
        // B layout (32-bit 4x16): lanes 0-15 hold rows (K0,K1) for col N=lane,
        // lanes 16-31 hold rows (K2,K3) for col N=lane-16.
        v2f b;
        b.x = lo ? ys : 1.0f;
        b.y = lo ? xs : 0.0f;
        v8f c = {};
        v8f d0 = __builtin_amdgcn_wmma_f32_16x16x4_f32(false, a0, false, b,
                                                       (short)0, c, false, false);
        v8f d1 = __builtin_amdgcn_wmma_f32_16x16x4_f32(false, a1, false, b,
                                                       (short)0, c, false, false);
        // in-lane argmin: this lane's columns are pixel l16; rows (centers) are
        // r + 8*(lane>=16) for d0 and +16 for d1, visited in increasing id order.
        int base = (lane >> 4) * 8;
        float bd = d0[0];
        int bi = base;
#pragma unroll
        for (int r = 1; r < 8; ++r) {
            if (d0[r] < bd) { bd = d0[r]; bi = base + r; }
        }
#pragma unroll
        for (int r = 0; r < 8; ++r) {
            if (d1[r] < bd) { bd = d1[r]; bi = base + 16 + r; }
        }
        // combine the two lane halves (same pixel, complementary center sets)
        float odv = __shfl_xor(bd, 16, 32);
        int   oiv = __shfl_xor(bi, 16, 32);
        if (odv < bd || (odv == bd && oiv < bi)) { bd = odv; bi = oiv; }
        if (lo) {
            int s = sem[p];
            int ns = (s >= 11 && anyv) ? s * 1000 + (bi + 1) : s;
            outSem[p] = ns;
        }
    }
}

// ---------------- k4: normals, camera heights, median keys ----------------
__device__ __forceinline__ void rayAt(float x, float y, const float* m,
                                      float& rx, float& ry, float& rz) {
    rx = fmaf(m[0], x, fmaf(m[1], y, m[2]));
    ry = fmaf(m[3], x, fmaf(m[4], y, m[5]));
    rz = fmaf(m[6], x, fmaf(m[7], y, m[8]));
}

__global__ void k4_height(const float* __restrict__ depth,
                          const float* __restrict__ Minv,
                          const float* __restrict__ out /*sem region*/,
                          char* ws) {
    __shared__ unsigned bc;
    if (threadIdx.x == 0) bc = 0u;
    __syncthreads();
    float m[9];
#pragma unroll
    for (int j = 0; j < 9; ++j) m[j] = Minv[j];
    const int* semOut = (const int*)out;
    unsigned* keys = (unsigned*)(ws + OFF_KEYS);
    unsigned myc = 0;
    int stride = gridDim.x * blockDim.x;
    for (int i = blockIdx.x * blockDim.x + threadIdx.x; i < HW_; i += stride) {
        int pf = i + stride;
        if (pf < HW_) __builtin_prefetch(&depth[pf], 0, 1);  // global_prefetch_b8
        int y = i >> 11, x = i & 2047;
        float dc = depth[i];
        float px, py, pz;
        rayAt((float)x, (float)y, m, px, py, pz);
        px *= dc; py *= dc; pz *= dc;
        int xm = (x > 0) ? x - 1 : 0;
        int xp = (x < W_ - 1) ? x + 1 : W_ - 1;
        int ym = (y > 0) ? y - 1 : 0;
        int yp = (y < H_ - 1) ? y + 1 : H_ - 1;
        float dxm = depth[(y << 11) | xm];
        float dxp = depth[(y << 11) | xp];
        float dym = depth[(ym << 11) | x];
        float dyp = depth[(yp << 11) | x];
        float ax, ay, az, bx, by, bz, tx, ty, tz;
        rayAt((float)xp, (float)y, m, ax, ay, az);
        rayAt((float)xm, (float)y, m, tx, ty, tz);
        float vxx = ax * dxp - tx * dxm;
        float vxy = ay * dxp - ty * dxm;
        float vxz = az * dxp - tz * dxm;
        rayAt((float)x, (float)yp, m, bx, by, bz);
        rayAt((float)x, (float)ym, m, tx, ty, tz);
        float vyx = bx * dyp - tx * dym;
        float vyy = by * dyp - ty * dym;
        float vyz = bz * dyp - tz * dym;
        // n = vx x vy
        float nx = vxy * vyz - vxz * vyy;
        float ny = vxz * vyx - vxx * vyz;
        float nz = vxx * vyy - vxy * vyx;
        float nn = sqrtf(nx * nx + ny * ny + nz * nz);
        float inv = 1.0f / (nn + 1e-8f);
        float h = fabsf(px * nx + py * ny + pz * nz) * inv;
        bool ground = (semOut[i] == 0);
        keys[i] = ground ? __float_as_uint(h) : 0xFFFFFFFFu;
        if (ground) ++myc;
    }
    atomicAdd(&bc, myc);
    __syncthreads();
    if (threadIdx.x == 0) atomicAdd((unsigned*)(ws + OFF_GROUND_CNT), bc);
}

// ---------------- median radix-select (4 passes, MSB first) ----------------
__global__ void k5_selinit(char* ws) {
    unsigned cnt = *(unsigned*)(ws + OFF_GROUND_CNT);
    int k = (cnt > 0) ? (int)((cnt - 1) / 2) : 0;
    *(int*)(ws + OFF_KREMAIN) = k;
    *(unsigned*)(ws + OFF_PREFIX) = 0u;
}

__global__ void k6_hist(const char* wsr, char* ws, int pass) {
    __shared__ unsigned sb[256];
    sb[threadIdx.x] = 0u;
    __syncthreads();
    const unsigned* keys = (const unsigned*)(wsr + OFF_KEYS);
    unsigned prefix = *(const unsigned*)(wsr + OFF_PREFIX);
    int bshift = 24 - 8 * pass;
    int i = blockIdx.x * blockDim.x + threadIdx.x;
    for (; i < HW_; i += gridDim.x * blockDim.x) {
        unsigned key = keys[i];
        bool ok = (pass == 0) || (((key ^ prefix) >> (32 - 8 * pass)) == 0u);
        if (ok) atomicAdd(&sb[(key >> bshift) & 255u], 1u);
    }
    __syncthreads();
    unsigned v = sb[threadIdx.x];
    if (v) atomicAdd((unsigned*)(ws + OFF_BINS) + pass * 256 + threadIdx.x, v);
}

__global__ void k7_pick(char* ws, int pass) {
    const unsigned* bins = (const unsigned*)(ws + OFF_BINS) + pass * 256;
    int k = *(int*)(ws + OFF_KREMAIN);
    unsigned prefix = *(unsigned*)(ws + OFF_PREFIX);
    unsigned cum = 0;
    int bshift = 24 - 8 * pass;
    for (int v = 0; v < 256; ++v) {
        unsigned c = bins[v];
        if ((int)(cum + c) > k) {
            *(unsigned*)(ws + OFF_PREFIX) = prefix | ((unsigned)v << bshift);
            *(int*)(ws + OFF_KREMAIN) = k - (int)cum;
            return;
        }
        cum += c;
    }
}

// ---------------- final: scale, filter, outputs ----------------
__global__ void k_final(const float* __restrict__ depth,
                        const float* __restrict__ Minv,
                        const float* __restrict__ rch,
                        const char* __restrict__ ws,
                        float* out) {
    float m[9];
#pragma unroll
    for (int j = 0; j < 9; ++j) m[j] = Minv[j];
    float med = __uint_as_float(*(const unsigned*)(ws + OFF_PREFIX));
    float scale = rch[0] / med;
    const int* semOut = (const int*)out;
    float* depthOut = out + HW_;
    float* camOut = out + 2 * HW_;
    int i = blockIdx.x * blockDim.x + threadIdx.x;
    for (; i < HW_; i += gridDim.x * blockDim.x) {
        int y = i >> 11, x = i & 2047;
        int s = semOut[i];
        float dc = depth[i];
        float px, py, pz;
        rayAt((float)x, (float)y, m, px, py, pz);
        float ds = dc * scale;
        px *= ds; py *= ds; pz *= ds;
        bool filt = (s == 10) || (s == 19);
        depthOut[i] = filt ? 0.0f : ds;
        float4 cv;
        cv.x = px; cv.y = py; cv.z = pz; cv.w = (float)s;
        *(float4*)(camOut + (size_t)i * 4) = cv;
    }
}

extern "C" void kernel_launch(void* const* d_in, const int* in_sizes, int n_in,
                              void* d_out, int out_size, void* d_ws, size_t ws_size,
                              hipStream_t stream) {
    const int*   sem   = (const int*)d_in[0];
    const float* heat  = (const float*)d_in[1];
    const float* off   = (const float*)d_in[2];
    const float* depth = (const float*)d_in[3];
    const float* Minv  = (const float*)d_in[4];
    const float* rch   = (const float*)d_in[5];
    float* out = (float*)d_out;
    char*  ws  = (char*)d_ws;

    (void)in_sizes; (void)n_in; (void)out_size; (void)ws_size;

    k0_zero<<<(ZERO_WORDS + 255) / 256, 256, 0, stream>>>((unsigned*)ws);
    k1_cand<<<1024, 256, 0, stream>>>(heat, ws);
    k2_top<<<1, 256, 0, stream>>>(ws);
    k3_inst<<<512, 256, 0, stream>>>(sem, off, ws, out);
    k4_height<<<1024, 256, 0, stream>>>(depth, Minv, out, ws);
    k5_selinit<<<1, 1, 0, stream>>>(ws);
    for (int pass = 0; pass < 4; ++pass) {
        k6_hist<<<1024, 256, 0, stream>>>(ws, ws, pass);
        k7_pick<<<1, 1, 0, stream>>>(ws, pass);
    }
    k_final<<<1024, 256, 0, stream>>>(depth, Minv, rch, ws, out);
}